// QLSTM_65481071406086
// MI455X (gfx1250) — compile-verified
//
#include <hip/hip_runtime.h>

typedef __attribute__((ext_vector_type(2))) float v2f;
typedef __attribute__((ext_vector_type(8))) float v8f;

// Fast device math: angles/cell values are analytically bounded (|arg| < ~40),
// so hardware v_cos_f32 / v_exp_f32 are accurate here; avoids libm's
// Payne-Hanek slow path (huge VALU/SALU + divergent branches).
__device__ __forceinline__ float fsigm(float x) { return 1.f / (1.f + __expf(-x)); }
__device__ __forceinline__ float ftanh(float x) {
  float t = __expf(2.f * x);
  return (t - 1.f) / (t + 1.f);
}

// Block = 128 threads (4 waves) <-> 32 batch rows, full T-loop inside
// (recurrence is per-row: no global sync needed anywhere).
// Per step:
//   phase 1 (all 4 waves): chained V_WMMA_F32_16X16X4_F32 (6 x K4 = K24, exact fp32)
//     computes a 16(row)x16(col) tile of comb(32x24) @ Wcat^T(24x32) + bias.
//     wave w: row-group w>>1, N-tile w&1.  D -> ang_sh.
//   phase 2 (thread = (row=lane, gate=wave)): closed-form qlayer
//     (RX angle-addition + Heisenberg conjugation of Z_w through the CNOT
//      chain; all Y-terms carry an X factor whose product-state expectation
//      is 0):
//     z_w  = cos(th_w) * prod_{v<=w} cos(a_v + th_v)      (w = 0..6)
//     z_7  = prod_{v<=6} cos(a_v + th_v) * cos(a_7 + 2*th_7)
//     then gate nonlinearity -> zz_sh.
//   phase 3 (thread = row, 32 threads): LSTM cell update, hx -> LDS + out.
__global__ __launch_bounds__(128) void qlstm_kernel(
    const float* __restrict__ x,
    const float* __restrict__ Wf, const float* __restrict__ bf,
    const float* __restrict__ Wi, const float* __restrict__ bi,
    const float* __restrict__ Wu, const float* __restrict__ bu,
    const float* __restrict__ Wo, const float* __restrict__ bo,
    const float* __restrict__ thf, const float* __restrict__ thi,
    const float* __restrict__ thu, const float* __restrict__ tho,
    float* __restrict__ out, int T, int B)
{
  __shared__ float hx_sh[32][8];
  __shared__ float ang_sh[32][33];   // [row][gate*8+wire], padded (33) vs bank conflicts
  __shared__ float zz_sh[32][33];    // activated gate values

  const int tid  = threadIdx.x;
  const int lane = tid & 31;
  const int wv   = __builtin_amdgcn_readfirstlane(tid >> 5);  // 0..3 (scalar)
  const int rg   = wv >> 1;          // row group (0/1)
  const int tile = wv & 1;           // N tile (0/1)
  const int r0   = blockIdx.x * 32;
  const int L    = lane & 15;
  const int half = lane >> 4;

  #pragma unroll
  for (int i = tid; i < 32 * 8; i += 128) ((float*)hx_sh)[i] = 0.f;

  // ---- loop-invariant, wave-uniform theta data (wave == gate in phase 2) ---
  const float* thg = (wv == 0) ? thf : ((wv == 1) ? thi : ((wv == 2) ? thu : tho));
  float tang[8];
  #pragma unroll
  for (int w = 0; w < 8; ++w) tang[w] = thg[w];
  float cth[7];
  #pragma unroll
  for (int w = 0; w < 7; ++w) cth[w] = __cosf(tang[w]);
  const float th7x2 = 2.f * tang[7];

  // ---- persistent WMMA B chunks (24 = 6 x K4) + bias accumulator C --------
  // f32 16x16x4 A layout: lanes0-15 M=lane {K0,K1}; lanes16-31 M=lane-16 {K2,K3}.
  // B mirrors with N<->M: lane holds W[N][K0..K0+1], K0 = 4*kc + 2*half.
  v2f Bk[6];
  v8f Cb;
  {
    const float* Wg = (tile == 0) ? ((L < 8) ? Wf : Wi) : ((L < 8) ? Wu : Wo);
    const float* bg = (tile == 0) ? ((L < 8) ? bf : bi) : ((L < 8) ? bu : bo);
    const float* Wrow = Wg + (L & 7) * 24;
    #pragma unroll
    for (int kc = 0; kc < 6; ++kc) {
      const int K0 = 4 * kc + 2 * half;
      Bk[kc].x = Wrow[K0];
      Bk[kc].y = Wrow[K0 + 1];
    }
    const float bv = bg[L & 7];
    #pragma unroll
    for (int v = 0; v < 8; ++v) Cb[v] = bv;   // C[M][N] = bias[N] for all M
  }

  float cx[8], hxv[8];                        // live in threads tid<32 (phase 3)
  #pragma unroll
  for (int k = 0; k < 8; ++k) { cx[k] = 0.f; hxv[k] = 0.f; }

  const int   gemm_row = r0 + rg * 16 + L;    // row this lane feeds in phase 1
  const size_t TB8 = (size_t)T * B * 8;

  for (int t = 0; t < T; ++t) {
    __syncthreads();                          // hx_sh ready / ang_sh consumed

    { // ---- phase 1: GEMM via chained fp32 WMMA --------------------------
      const float* xr = x + ((size_t)t * B + gemm_row) * 16;
      if (t + 1 < T) __builtin_prefetch(xr + (size_t)B * 16, 0, 3); // next step

      v8f acc = Cb;
      #pragma unroll
      for (int kc = 0; kc < 4; ++kc) {        // K = 0..15 from x_t
        v2f a;
        const int K0 = 4 * kc + 2 * half;
        a.x = xr[K0];
        a.y = xr[K0 + 1];
        acc = __builtin_amdgcn_wmma_f32_16x16x4_f32(false, a, false, Bk[kc],
                                                    (short)0, acc, false, false);
      }
      #pragma unroll
      for (int kc = 4; kc < 6; ++kc) {        // K = 16..23 from hx (branch-free)
        v2f a;
        const int K0 = 4 * (kc - 4) + 2 * half;
        a.x = hx_sh[rg * 16 + L][K0];
        a.y = hx_sh[rg * 16 + L][K0 + 1];
        acc = __builtin_amdgcn_wmma_f32_16x16x4_f32(false, a, false, Bk[kc],
                                                    (short)0, acc, false, false);
      }
      #pragma unroll
      for (int v = 0; v < 8; ++v)             // D: vgpr v -> M = v (+8 for hi half)
        ang_sh[rg * 16 + v + half * 8][tile * 16 + L] = acc[v];
    }
    __syncthreads();                          // angles ready

    { // ---- phase 2: closed-form qlayer; thread = (row=lane, gate=wv) ----
      const float* ap = &ang_sh[lane][wv * 8];
      float P = 1.f;
      float g8[8];
      #pragma unroll
      for (int w = 0; w < 7; ++w) {
        P *= __cosf(ap[w] + tang[w]);         // prefix prod of cos(alpha_v)
        g8[w] = cth[w] * P;
      }
      g8[7] = P * __cosf(ap[7] + th7x2);
      #pragma unroll
      for (int w = 0; w < 8; ++w)
        zz_sh[lane][wv * 8 + w] = (wv == 2) ? ftanh(g8[w]) : fsigm(g8[w]);
    }
    __syncthreads();                          // gate values ready

    if (tid < 32) {                           // ---- phase 3: LSTM combine ----
      #pragma unroll
      for (int k = 0; k < 8; ++k) {
        const float f = zz_sh[tid][k];
        const float i = zz_sh[tid][8 + k];
        const float g = zz_sh[tid][16 + k];
        const float o = zz_sh[tid][24 + k];
        cx[k]  = fmaf(f, cx[k], i * g);
        hxv[k] = o * ftanh(cx[k]);
        hx_sh[tid][k] = hxv[k];
      }
      float4* op = (float4*)(out + ((size_t)t * B + (r0 + tid)) * 8);
      op[0] = make_float4(hxv[0], hxv[1], hxv[2], hxv[3]);
      op[1] = make_float4(hxv[4], hxv[5], hxv[6], hxv[7]);
    }
  }

  if (tid < 32) {                             // final hx, cx tails
    float4* hp = (float4*)(out + TB8 + (size_t)(r0 + tid) * 8);
    hp[0] = make_float4(hxv[0], hxv[1], hxv[2], hxv[3]);
    hp[1] = make_float4(hxv[4], hxv[5], hxv[6], hxv[7]);
    float4* cp = (float4*)(out + TB8 + (size_t)B * 8 + (size_t)(r0 + tid) * 8);
    cp[0] = make_float4(cx[0], cx[1], cx[2], cx[3]);
    cp[1] = make_float4(cx[4], cx[5], cx[6], cx[7]);
  }
}

extern "C" void kernel_launch(void* const* d_in, const int* in_sizes, int n_in,
                              void* d_out, int out_size, void* d_ws, size_t ws_size,
                              hipStream_t stream) {
  (void)n_in; (void)out_size; (void)d_ws; (void)ws_size;
  const float* x   = (const float*)d_in[0];
  const float* Wf  = (const float*)d_in[1];
  const float* bf  = (const float*)d_in[2];
  const float* Wi  = (const float*)d_in[3];
  const float* bi  = (const float*)d_in[4];
  const float* Wu  = (const float*)d_in[5];
  const float* bu  = (const float*)d_in[6];
  const float* Wo  = (const float*)d_in[7];
  const float* bo  = (const float*)d_in[8];
  const float* thf = (const float*)d_in[9];
  const float* thi = (const float*)d_in[10];
  const float* thu = (const float*)d_in[11];
  const float* tho = (const float*)d_in[12];

  const int T = 256;
  const int B = in_sizes[0] / (T * 16);   // 8192
  dim3 grid(B / 32), block(128);
  qlstm_kernel<<<grid, block, 0, stream>>>(x, Wf, bf, Wi, bi, Wu, bu, Wo, bo,
                                           thf, thi, thu, tho,
                                           (float*)d_out, T, B);
}